// GCNLayer_28106265985527
// MI455X (gfx1250) — compile-verified
//
#include <hip/hip_runtime.h>

typedef float v2f __attribute__((ext_vector_type(2)));
typedef float v8f __attribute__((ext_vector_type(8)));

#define N_NODES   100000
#define N_EDGES   1600000
#define IN_FEATS  256
#define OUT_FEATS 128
#define ROW_TILES (N_NODES / 16)   // 6250 (N_NODES divisible by 16)

// -------- GEMM: support[N,128] = inputs[N,256] @ W[256,128], f32 WMMA ------
__global__ __launch_bounds__(256) void gcn_gemm_wmma(
    const float* __restrict__ A, const float* __restrict__ W,
    float* __restrict__ S)
{
    // W staged in LDS, pre-swizzled into WMMA B-fragment order:
    // element W[kk][n]:  kstep=kk>>2, kin=kk&3, c=n>>4, nin=n&15
    //   lane = (kin>>1)*16 + nin, j = kin&1
    //   lds_idx = kstep*512 + c*64 + lane*2 + j
    // => per (kstep,c), lane L reads a contiguous float2 at lane*2: one
    //    conflict-free ds_load_b64 per B fragment.
    __shared__ float Wlds[IN_FEATS * OUT_FEATS];   // 128 KB of 320 KB WGP LDS

    const int t = threadIdx.x;
    #pragma unroll
    for (int i = 0; i < (IN_FEATS * OUT_FEATS) / 256; ++i) {
        int idx = i * 256 + t;            // coalesced global read of W
        int kk  = idx >> 7;               // row (K index)
        int n   = idx & 127;              // col (N index)
        float w = W[idx];
        int kstep = kk >> 2, kin = kk & 3;
        int c = n >> 4, nin = n & 15;
        int lane = ((kin >> 1) << 4) + nin;
        int j = kin & 1;
        Wlds[kstep * 512 + c * 64 + lane * 2 + j] = w;
    }
    __syncthreads();

    const int wave = t >> 5;
    const int lane = t & 31;
    int tile = blockIdx.x * 8 + wave;                 // 16-row tile index
    if (tile > ROW_TILES - 1) tile = ROW_TILES - 1;   // clamp: EXEC stays all-1s
    const int r0    = tile * 16;
    const int m     = lane & 15;
    const int khalf = (lane >> 4) << 1;               // lanes 16-31 hold K=2,3

    v8f acc[8];
    #pragma unroll
    for (int c = 0; c < 8; ++c) acc[c] = (v8f)0.0f;

    const float* arow = A + (size_t)(r0 + m) * IN_FEATS + khalf;

    for (int ks = 0; ks < IN_FEATS / 4; ++ks) {
        // A fragment: 16x4 f32, streamed once -> non-temporal
        v2f a = __builtin_nontemporal_load((const v2f*)(arow + ks * 4));
        const float* wl = &Wlds[ks * 512 + lane * 2];
        #pragma unroll
        for (int c = 0; c < 8; ++c) {
            v2f b = *(const v2f*)(wl + c * 64);
            acc[c] = __builtin_amdgcn_wmma_f32_16x16x4_f32(
                false, a, false, b, (short)0, acc[c], false, false);
        }
    }

    // D layout: VGPR i -> lanes 0-15: M=i, lanes 16-31: M=8+i; col = lane&15
    const int mrow = (lane >> 4) << 3;
    const int col  = lane & 15;
    #pragma unroll
    for (int c = 0; c < 8; ++c) {
        #pragma unroll
        for (int i = 0; i < 8; ++i) {
            int row = r0 + mrow + i;
            S[(size_t)row * OUT_FEATS + c * 16 + col] = acc[c][i];
        }
    }
}

// -------- init: out[n][f] = B[f] (harness poisons d_out) -------------------
__global__ void gcn_init_out(float* __restrict__ out, const float* __restrict__ B)
{
    int i = blockIdx.x * blockDim.x + threadIdx.x;
    if (i < N_NODES * OUT_FEATS) out[i] = B[i & (OUT_FEATS - 1)];
}

// -------- scatter: out[dst] += w_e * support[src]; wave-per-edge -----------
__global__ __launch_bounds__(256) void gcn_scatter(
    const float* __restrict__ S, const int* __restrict__ esrc,
    const int* __restrict__ edst, const float* __restrict__ ew,
    float* __restrict__ out)
{
    const int gid  = blockIdx.x * blockDim.x + threadIdx.x;
    const int e    = gid >> 5;           // one wave32 per edge
    if (e >= N_EDGES) return;
    const int lane = gid & 31;           // 4 feats per lane -> 128 feats

    const int   src = esrc[e];
    const int   dst = edst[e];
    const float w   = ew[e];

    // coalesced 512B row read; support row lives in L2 (51.2 MB < 192 MB)
    const float4 v = *(const float4*)(S + (size_t)src * OUT_FEATS + lane * 4);
    float* o = out + (size_t)dst * OUT_FEATS + lane * 4;
    atomicAdd(o + 0, w * v.x);
    atomicAdd(o + 1, w * v.y);
    atomicAdd(o + 2, w * v.z);
    atomicAdd(o + 3, w * v.w);
}

extern "C" void kernel_launch(void* const* d_in, const int* in_sizes, int n_in,
                              void* d_out, int out_size, void* d_ws, size_t ws_size,
                              hipStream_t stream)
{
    const float* inputs = (const float*)d_in[0];
    const float* W      = (const float*)d_in[1];
    const float* B      = (const float*)d_in[2];
    const int*   esrc   = (const int*)d_in[3];
    const int*   edst   = (const int*)d_in[4];
    const float* ew     = (const float*)d_in[5];
    float*       out    = (float*)d_out;
    float*       S      = (float*)d_ws;   // support: 100000*128 f32 = 51.2 MB

    // 6250 row tiles, 8 waves/block -> 782 blocks (tail clamped in-kernel)
    gcn_gemm_wmma<<<(ROW_TILES + 7) / 8, 256, 0, stream>>>(inputs, W, S);
    gcn_init_out<<<(N_NODES * OUT_FEATS + 255) / 256, 256, 0, stream>>>(out, B);
    gcn_scatter<<<(N_EDGES * 32) / 256, 256, 0, stream>>>(S, esrc, edst, ew, out);
}